// SparseBiasDiagUnfolder_76459007803909
// MI455X (gfx1250) — compile-verified
//
#include <hip/hip_runtime.h>
#include <cstdint>

// Problem constants fixed by setup_inputs():
//   adj: (B=2, N=2048, N=2048, F=16) f32, filter=8, stride=4, n=2048
//   starts = 0,4,...,2040  -> S = 511 windows
//   off-diagonal positions per window: P = 8*8 - 8 = 56
//   out: (B, S, P*F) = (2, 511, 896) f32
namespace {
constexpr int kB       = 2;
constexpr int kN       = 2048;
constexpr int kF       = 16;                      // channels (64 bytes contiguous)
constexpr int kFilt    = 8;
constexpr int kStride  = 4;
constexpr int kS       = (kN - kFilt) / kStride + 1;   // 511
constexpr int kP       = kFilt * kFilt - kFilt;        // 56
constexpr int kThreads = kB * kP * 4;                  // 448 = 14 wave32s; one b128 (4 floats) per lane
constexpr unsigned kLdsBytes = (unsigned)kThreads * 16u; // 7168 B dynamic LDS staging
}

__global__ __launch_bounds__(kThreads)
void offdiag_unfold_async(const float* __restrict__ adj, float* __restrict__ out) {
  const int tid = threadIdx.x;          // 0..447, all lanes active (full EXEC)
  const int s   = blockIdx.x;           // 0..510  (window index)
  const int c4  = tid & 3;              // which 16B quarter of the 64B channel chunk
  const int p   = (tid >> 2) % kP;      // packed off-diagonal position 0..55
  const int b   = tid / (kP * 4);       // batch 0..1

  // Closed-form off-diagonal mapping: l-th kept cell of the 8x8 window,
  // skipping the diagonal (l % 9 == 0):  l = p + 1 + p/8
  const int l   = p + 1 + (p >> 3);
  const int ii  = l >> 3;
  const int jj  = l & 7;

  const int row = s * kStride + ii;
  const int col = s * kStride + jj;

  // Flat f32 element indices (max 134,217,727 < 2^31 -> 32-bit safe)
  const int srcIdx = ((b * kN + row) * kN + col) * kF + c4 * 4;
  const int dstIdx = ((b * kS + s) * kP + p) * kF + c4 * 4;

  const unsigned long long srcAddr = (unsigned long long)(uintptr_t)(adj + srcIdx);
  const unsigned long long dstAddr = (unsigned long long)(uintptr_t)(out + dstIdx);

  // Per-lane LDS staging slot. No static LDS is declared, so the dynamic LDS
  // region begins at byte offset 0 of this workgroup's allocation.
  const unsigned ldsAddr = (unsigned)tid * 16u;

  // CDNA5 async data movement: global -> LDS -> global, data never in VGPRs.
  // Tracked by ASYNCcnt; load completion and store's LDS read are unordered,
  // so we must drain ASYNCcnt between them (CDNA5 ISA).
  asm volatile("global_load_async_to_lds_b128 %0, %1, off"
               :: "v"(ldsAddr), "v"(srcAddr)
               : "memory");
  asm volatile("s_wait_asynccnt 0" ::: "memory");
  asm volatile("global_store_async_from_lds_b128 %0, %1, off"
               :: "v"(dstAddr), "v"(ldsAddr)
               : "memory");
  // S_ENDPGM performs an implicit wait-idle, but drain explicitly anyway.
  asm volatile("s_wait_asynccnt 0" ::: "memory");
}

extern "C" void kernel_launch(void* const* d_in, const int* in_sizes, int n_in,
                              void* d_out, int out_size, void* d_ws, size_t ws_size,
                              hipStream_t stream) {
  (void)in_sizes; (void)n_in; (void)d_ws; (void)ws_size; (void)out_size;
  const float* adj = (const float*)d_in[0];   // f32 input tensor
  float*       out = (float*)d_out;           // f32 output
  hipLaunchKernelGGL(offdiag_unfold_async,
                     dim3(kS), dim3(kThreads), kLdsBytes, stream,
                     adj, out);
}